// model003a_37666863186446
// MI455X (gfx1250) — compile-verified
//
#include <hip/hip_runtime.h>

#define NY 64
#define NU 64
#define NH 256
#define T_STEPS 4096
#define NB 32
#define NCH 32                 // number of scan chunks
#define CHUNK (T_STEPS / NCH)  // 128 timesteps per chunk

typedef float v2f __attribute__((ext_vector_type(2)));
typedef float v8f __attribute__((ext_vector_type(8)));

// ---------------------------------------------------------------------------
// Generic fp32 WMMA GEMM: C[M,N] = A[M,K] @ Bt[N,K]^T (+ bias[N])
// One 16x16 output tile per wave32, K-loop of V_WMMA_F32_16X16X4_F32.
//
// Fragment layouts per CDNA5 ISA (CDNA5 ISA):
//  A 16x4 f32 : lanes 0-15 -> M=lane,    VGPR0=K0, VGPR1=K1
//               lanes 16-31 -> M=lane-16, VGPR0=K2, VGPR1=K3
//  B 4x16 f32 : lanes 0-15 -> N=lane,    VGPR0=K0, VGPR1=K1
//               lanes 16-31 -> N=lane-16, VGPR0=K2, VGPR1=K3
//  C 16x16 f32: VGPR i -> M = i (lanes 0-15) / i+8 (lanes 16-31), N = lane&15
// ---------------------------------------------------------------------------
__global__ void gemm_f32_wmma(const float* __restrict__ A,
                              const float* __restrict__ Bt,
                              const float* __restrict__ bias,
                              float* __restrict__ C,
                              int M, int N, int K) {
  const int lane   = threadIdx.x;        // 0..31
  const int r      = lane & 15;
  const int kb     = (lane >> 4) << 1;   // 0 for lanes 0-15, 2 for lanes 16-31
  const int tilesN = N >> 4;
  const int waveId = blockIdx.x * blockDim.y + threadIdx.y;
  const int tileM  = (waveId / tilesN) << 4;
  const int tileN  = (waveId % tilesN) << 4;
  if (tileM >= M) return;  // wave-uniform; EXEC stays all-ones for WMMA

  const float* arow = A  + (size_t)(tileM + r) * K + kb;
  const float* brow = Bt + (size_t)(tileN + r) * K + kb;

  v8f acc = {};
  for (int k0 = 0; k0 < K; k0 += 4) {
    v2f afrag = *(const v2f*)(arow + k0);   // 8B-aligned: k0%4==0, kb even
    v2f bfrag = *(const v2f*)(brow + k0);
    // (neg_a, A, neg_b, B, c_mod, C, reuse_a, reuse_b)
    acc = __builtin_amdgcn_wmma_f32_16x16x4_f32(false, afrag, false, bfrag,
                                                (short)0, acc, false, false);
  }

  if (bias != nullptr) {
    const float bv = bias[tileN + r];
#pragma unroll
    for (int i = 0; i < 8; ++i) acc[i] += bv;
  }

  const int rowBase = tileM + ((lane >> 4) << 3);
#pragma unroll
  for (int i = 0; i < 8; ++i) {
    C[(size_t)(rowBase + i) * N + (tileN + r)] = acc[i];
  }
}

// ---------------------------------------------------------------------------
// Chunked parallel scan for z_t = lambda * z_{t-1} + u_t  (lambda const per h).
// Serial depth 4096 -> 128 + 32 + 128; phases A/C run 4096 waves.
// ---------------------------------------------------------------------------
__device__ __forceinline__ void load_lambda(const float* lre, const float* lim,
                                            int h, float& lr, float& li) {
  const float rr = __expf(-__builtin_fabsf(lre[h]));
  const float th = 1.57079632679489662f * lim[h];  // pi/2 * imag
  lr = rr * __cosf(th);
  li = rr * __sinf(th);
}

// Phase A: per (chunk c, batch b, channel h) compute the chunk-local input
// response L_c = sum_j lambda^{CHUNK-1-j} * u_{c*CHUNK+j}  (iterate from z=0).
__global__ void scan_chunk_reduce(const float* __restrict__ Bu,
                                  float2* __restrict__ S,      // [NCH][NB][NH/2]
                                  const float* __restrict__ lre,
                                  const float* __restrict__ lim) {
  const int id = blockIdx.x * blockDim.x + threadIdx.x;  // (c*NB + b)*128 + h
  const int h  = id & 127;
  const int b  = (id >> 7) & (NB - 1);
  const int c  = id >> 12;

  float lr, li;
  load_lambda(lre, lim, h, lr, li);

  float zr = 0.0f, zi = 0.0f;
  const float* p = Bu + ((size_t)(c * CHUNK) * NB + b) * NH + h;
  const size_t stride = (size_t)NB * NH;
#pragma unroll 4
  for (int j = 0; j < CHUNK; ++j) {
    const float ak = p[0];
    const float bk = p[NH / 2];
    const float nzr = fmaf(lr, zr, fmaf(-li, zi, ak));
    const float nzi = fmaf(li, zr, fmaf(lr, zi, bk));
    zr = nzr; zi = nzi;
    p += stride;
  }
  S[id] = make_float2(zr, zi);
}

// Phase B: sequential combine over the 32 chunk summaries (per (b,h)):
//   z_start[0] = x0 ;  z_start[c+1] = lambda^CHUNK * z_start[c] + L_c
__global__ void scan_combine(const float2* __restrict__ S,   // [NCH][NB][NH/2]
                             float2* __restrict__ Zs,        // [NCH][NB][NH/2]
                             const float* __restrict__ x0,   // [NB][NH]
                             const float* __restrict__ lre,
                             const float* __restrict__ lim) {
  const int id = blockIdx.x * blockDim.x + threadIdx.x;  // b*128 + h
  const int h  = id & 127;
  const int b  = id >> 7;

  float lr, li;
  load_lambda(lre, lim, h, lr, li);

  // lambda^CHUNK via log2(CHUNK)=7 complex squarings
  float ar = lr, ai = li;
#pragma unroll
  for (int i = 0; i < 7; ++i) {
    const float nr = ar * ar - ai * ai;
    const float ni = 2.0f * ar * ai;
    ar = nr; ai = ni;
  }

  float zr = x0[b * NH + h];
  float zi = x0[b * NH + (NH / 2) + h];
#pragma unroll
  for (int c = 0; c < NCH; ++c) {
    const int idx = (c * NB + b) * (NH / 2) + h;
    Zs[idx] = make_float2(zr, zi);
    const float2 s = S[idx];
    const float nzr = fmaf(ar, zr, fmaf(-ai, zi, s.x));
    const float nzi = fmaf(ai, zr, fmaf(ar, zi, s.y));
    zr = nzr; zi = nzi;
  }
}

// Phase C: re-run each chunk from its exact start state, writing X in place
// over Bu (each thread owns its (b,h) addresses; Bu is L2-resident at 134 MB).
__global__ void scan_chunk_apply(float* __restrict__ Bu,
                                 const float2* __restrict__ Zs,
                                 const float* __restrict__ lre,
                                 const float* __restrict__ lim) {
  const int id = blockIdx.x * blockDim.x + threadIdx.x;  // (c*NB + b)*128 + h
  const int h  = id & 127;
  const int b  = (id >> 7) & (NB - 1);
  const int c  = id >> 12;

  float lr, li;
  load_lambda(lre, lim, h, lr, li);

  const float2 z0 = Zs[id];
  float zr = z0.x, zi = z0.y;

  float* p = Bu + ((size_t)(c * CHUNK) * NB + b) * NH + h;
  const size_t stride = (size_t)NB * NH;
#pragma unroll 4
  for (int j = 0; j < CHUNK; ++j) {
    const float ak = p[0];
    const float bk = p[NH / 2];
    const float nzr = fmaf(lr, zr, fmaf(-li, zi, ak));
    const float nzi = fmaf(li, zr, fmaf(lr, zi, bk));
    p[0]      = nzr;
    p[NH / 2] = nzi;
    zr = nzr; zi = nzi;
    p += stride;
  }
}

extern "C" void kernel_launch(void* const* d_in, const int* in_sizes, int n_in,
                              void* d_out, int out_size, void* d_ws, size_t ws_size,
                              hipStream_t stream) {
  (void)in_sizes; (void)n_in; (void)out_size; (void)ws_size;

  const float* y0   = (const float*)d_in[0];  // [NB, NY]
  const float* U    = (const float*)d_in[1];  // [T, NB, NU]
  const float* lre  = (const float*)d_in[2];  // [NH/2]
  const float* lim  = (const float*)d_in[3];  // [NH/2]
  const float* Bm   = (const float*)d_in[4];  // [NH, NU]
  const float* Wy2x = (const float*)d_in[5];  // [NH, NY]
  const float* by2x = (const float*)d_in[6];  // [NH]
  const float* Wx2y = (const float*)d_in[7];  // [NY, NH]
  const float* bx2y = (const float*)d_in[8];  // [NY]
  float* Y = (float*)d_out;                   // [T, NB, NY]

  // Workspace layout
  char* ws = (char*)d_ws;
  float* Bu = (float*)ws;  // T*NB*NH f32  (134 MB; also holds X in place)
  size_t off = (size_t)T_STEPS * NB * NH * sizeof(float);
  float* x0 = (float*)(ws + off);  // NB*NH f32
  off += (size_t)NB * NH * sizeof(float);
  float2* S = (float2*)(ws + off);  // NCH*NB*(NH/2) float2  (1 MB)
  off += (size_t)NCH * NB * (NH / 2) * sizeof(float2);
  float2* Zs = (float2*)(ws + off);  // NCH*NB*(NH/2) float2 (1 MB)

  const int MB = T_STEPS * NB;  // 131072 rows for the big GEMMs
  const dim3 blk(32, 8);        // 8 wave32s per block

  // 1) Bu[t,b,h] = sum_u U[t,b,u] * B[h,u]   -> (131072 x 64) @ (64 x 256)^T
  {
    const int waves = (MB / 16) * (NH / 16);
    gemm_f32_wmma<<<waves / 8, blk, 0, stream>>>(U, Bm, nullptr, Bu, MB, NH, NU);
  }

  // 2) x0 = y0 @ W_y2x^T + b_y2x             -> (32 x 64) @ (64 x 256)^T
  {
    const int waves = (NB / 16) * (NH / 16);  // 32 waves
    gemm_f32_wmma<<<waves / 8, blk, 0, stream>>>(y0, Wy2x, by2x, x0, NB, NH, NU);
  }

  // 3) Chunked parallel scan (A: local reduce, B: combine, C: apply in place)
  {
    const int nA = NCH * NB * (NH / 2);  // 131072 threads
    scan_chunk_reduce<<<nA / 256, 256, 0, stream>>>(Bu, S, lre, lim);

    const int nB = NB * (NH / 2);        // 4096 threads
    scan_combine<<<nB / 256, 256, 0, stream>>>(S, Zs, x0, lre, lim);

    scan_chunk_apply<<<nA / 256, 256, 0, stream>>>(Bu, Zs, lre, lim);
  }

  // 4) Y = X @ W_x2y^T + b_x2y               -> (131072 x 256) @ (256 x 64)^T
  {
    const int waves = (MB / 16) * (NY / 16);
    gemm_f32_wmma<<<waves / 8, blk, 0, stream>>>(Bu, Wx2y, bx2y, Y, MB, NY, NH);
  }
}